// RecSysGNN_88596585382821
// MI455X (gfx1250) — compile-verified
//
#include <hip/hip_runtime.h>
#include <stdint.h>

#define LGC_D 64

// vector-of-2-int type matching the builtin's parameter type
typedef int lgc_v2i __attribute__((vector_size(2 * sizeof(int))));
typedef __attribute__((address_space(1))) lgc_v2i* lgc_gv2i_ptr;
typedef __attribute__((address_space(3))) lgc_v2i* lgc_lv2i_ptr;

// ---- CDNA5 async global->LDS path (probe via __has_builtin; fallback = shfl) ----
#if defined(__HIP_DEVICE_COMPILE__) && __has_builtin(__builtin_amdgcn_global_load_async_to_lds_b64)
#define LGC_ASYNC 1
#else
#define LGC_ASYNC 0
#endif

__device__ __forceinline__ void lgc_wait_async0() {
#if __has_builtin(__builtin_amdgcn_s_wait_asynccnt)
  __builtin_amdgcn_s_wait_asynccnt(0);
#else
  asm volatile("s_wait_asynccnt 0" ::: "memory");
#endif
}
__device__ __forceinline__ void lgc_wait_async1() {
#if __has_builtin(__builtin_amdgcn_s_wait_asynccnt)
  __builtin_amdgcn_s_wait_asynccnt(1);
#else
  asm volatile("s_wait_asynccnt 1" ::: "memory");
#endif
}

// ---------------- setup kernels ----------------

__global__ void lgc_count(const int* __restrict__ dst, unsigned* __restrict__ deg, int E) {
  int e = blockIdx.x * blockDim.x + threadIdx.x;
  if (e < E) atomicAdd(&deg[dst[e]], 1u);
}

__global__ void lgc_dinv(const unsigned* __restrict__ deg, float* __restrict__ dinv, int N) {
  int n = blockIdx.x * blockDim.x + threadIdx.x;
  if (n < N) {
    unsigned d = deg[n];
    dinv[n] = d ? (1.0f / sqrtf((float)d)) : 0.0f;
  }
}

// single-workgroup exclusive scan of deg -> row_ptr (and cursor copy)
__global__ __launch_bounds__(1024) void lgc_scan(const unsigned* __restrict__ deg,
                                                 int* __restrict__ row_ptr,
                                                 int* __restrict__ cursor, int N) {
  __shared__ int sums[1024];
  const int t = threadIdx.x;
  const int chunk = (N + 1023) / 1024;
  const int s0 = t * chunk;
  const int s1 = min(N, s0 + chunk);
  int local = 0;
  for (int i = s0; i < s1; ++i) local += (int)deg[i];
  sums[t] = local;
  __syncthreads();
  for (int off = 1; off < 1024; off <<= 1) {
    int v = (t >= off) ? sums[t - off] : 0;
    __syncthreads();
    sums[t] += v;
    __syncthreads();
  }
  int base = (t == 0) ? 0 : sums[t - 1];   // exclusive prefix
  for (int i = s0; i < s1; ++i) {
    row_ptr[i] = base;
    cursor[i]  = base;
    base += (int)deg[i];
  }
  if (t == 1023) row_ptr[N] = base;        // total = E
}

// fill CSR: packed 8B record per edge {hi: norm bits, lo: src}
__global__ void lgc_fill(const int* __restrict__ src, const int* __restrict__ dst,
                         const float* __restrict__ dinv, int* __restrict__ cursor,
                         long long* __restrict__ csr, int E) {
  int e = blockIdx.x * blockDim.x + threadIdx.x;
  if (e >= E) return;
  int s = src[e], d = dst[e];
  int pos = atomicAdd(&cursor[d], 1);
  float w = dinv[s] * dinv[d];
  unsigned long long rec =
      ((unsigned long long)(unsigned)__float_as_int(w) << 32) | (unsigned)s;
  csr[pos] = (long long)rec;
}

// ---------------- main propagation: one wave per dst node ----------------
// lane holds columns [2*lane, 2*lane+1]; edge records staged into LDS via
// double-buffered async global->LDS loads, then broadcast-read per edge.
__global__ __launch_bounds__(256) void lgc_propagate(
    const int* __restrict__ row_ptr,
    const long long* __restrict__ csr,
    const float* __restrict__ emb_in,
    float* __restrict__ emb_out,
    const float* __restrict__ emb0,
    float* __restrict__ acc,
    int N, int accInit) {
  const int lane = threadIdx.x & 31;
  const int wv   = threadIdx.x >> 5;
  const int node = blockIdx.x * 8 + wv;
  if (node >= N) return;

  const int beg = row_ptr[node];
  const int end = row_ptr[node + 1];

  float ax = 0.0f, ay = 0.0f;

#if LGC_ASYNC
  __shared__ long long stage[8][2][32];
  // prime buffer 0 with the first chunk
  if (beg < end) {
    const int c0 = min(32, end - beg);
    if (lane < c0)
      __builtin_amdgcn_global_load_async_to_lds_b64(
          (lgc_gv2i_ptr)(csr + beg + lane),
          (lgc_lv2i_ptr)&stage[wv][0][lane], 0, 0);
  }
  int buf = 0;
  for (int base = beg; base < end; base += 32, buf ^= 1) {
    const int cnt = min(32, end - base);
    const int nxt = base + 32;
    if (base + 64 + lane < end)               // prefetch chunk after next
      __builtin_prefetch(csr + base + 64 + lane, 0, 0);
    if (nxt < end) {
      const int cn = min(32, end - nxt);
      if (lane < cn)                          // stage next chunk into other buffer
        __builtin_amdgcn_global_load_async_to_lds_b64(
            (lgc_gv2i_ptr)(csr + nxt + lane),
            (lgc_lv2i_ptr)&stage[wv][buf ^ 1][lane], 0, 0);
      lgc_wait_async1();                      // current chunk done, next in flight
    } else {
      lgc_wait_async0();
    }
    for (int j = 0; j < cnt; ++j) {
      const long long e = stage[wv][buf][j];  // LDS broadcast read
      const int   sj = (int)((unsigned long long)e & 0xffffffffull);
      const float wj = __int_as_float((int)((unsigned long long)e >> 32));
      const float2 v = *(const float2*)(emb_in + ((size_t)sj << 6) + lane * 2);
      ax = fmaf(wj, v.x, ax);
      ay = fmaf(wj, v.y, ay);
    }
  }
#else
  for (int base = beg; base < end; base += 32) {
    const int cnt = min(32, end - base);
    if (base + 32 + lane < end)
      __builtin_prefetch(csr + base + 32 + lane, 0, 0);
    long long ew = (lane < cnt) ? csr[base + lane] : 0ll;
    for (int j = 0; j < cnt; ++j) {
      const long long e = __shfl(ew, j, 32);
      const int   sj = (int)((unsigned long long)e & 0xffffffffull);
      const float wj = __int_as_float((int)((unsigned long long)e >> 32));
      const float2 v = *(const float2*)(emb_in + ((size_t)sj << 6) + lane * 2);
      ax = fmaf(wj, v.x, ax);
      ay = fmaf(wj, v.y, ay);
    }
  }
#endif

  const size_t o = ((size_t)node << 6) + lane * 2;
  emb_out[o]     = ax;
  emb_out[o + 1] = ay;
  if (accInit) { acc[o] = emb0[o] + ax;  acc[o + 1] = emb0[o + 1] + ay; }
  else         { acc[o] += ax;           acc[o + 1] += ay; }
}

__global__ void lgc_finalize(const float* __restrict__ emb0,
                             float* __restrict__ outFirst,
                             float* __restrict__ outSecond, long long n) {
  long long i = (long long)blockIdx.x * blockDim.x + threadIdx.x;
  if (i < n) {
    outFirst[i]  = emb0[i];        // tuple element 0: emb0
    outSecond[i] *= 0.25f;         // tuple element 1: acc / (L+1)
  }
}

// ---------------- launcher ----------------

extern "C" void kernel_launch(void* const* d_in, const int* in_sizes, int n_in,
                              void* d_out, int out_size, void* d_ws, size_t ws_size,
                              hipStream_t stream) {
  (void)n_in; (void)out_size; (void)ws_size;
  const int*   edges = (const int*)d_in[0];
  const float* emb0  = (const float*)d_in[1];
  const int E = in_sizes[0] / 2;
  const int N = in_sizes[1] / LGC_D;
  const int* src = edges;
  const int* dst = edges + E;

  // workspace partition (256B aligned)
  char* p = (char*)d_ws;
  auto take = [&](size_t bytes) -> char* {
    char* r = p; p += (bytes + 255) & ~(size_t)255; return r;
  };
  unsigned*  deg     = (unsigned*)take((size_t)N * 4);
  int*       row_ptr = (int*)take(((size_t)N + 1) * 4);
  int*       cursor  = (int*)take((size_t)N * 4);
  float*     dinv    = (float*)take((size_t)N * 4);
  long long* csr     = (long long*)take((size_t)E * 8);
  float*     embA    = (float*)take((size_t)N * LGC_D * 4);

  float* outFirst  = (float*)d_out;                         // scratch embB, final emb0
  float* outSecond = (float*)d_out + (size_t)N * LGC_D;     // accumulator, final out

  (void)hipMemsetAsync(deg, 0, (size_t)N * 4, stream);

  const int TB = 256;
  lgc_count<<<(E + TB - 1) / TB, TB, 0, stream>>>(dst, deg, E);
  lgc_dinv <<<(N + TB - 1) / TB, TB, 0, stream>>>(deg, dinv, N);
  lgc_scan <<<1, 1024, 0, stream>>>(deg, row_ptr, cursor, N);
  lgc_fill <<<(E + TB - 1) / TB, TB, 0, stream>>>(src, dst, dinv, cursor, csr, E);

  const int PB = (N + 7) / 8;   // one wave (32 lanes) per node, 8 waves/block
  // layer 1: in=emb0      -> embA     ; acc = emb0 + e1
  lgc_propagate<<<PB, TB, 0, stream>>>(row_ptr, csr, emb0,     embA,     emb0, outSecond, N, 1);
  // layer 2: in=embA      -> outFirst ; acc += e2
  lgc_propagate<<<PB, TB, 0, stream>>>(row_ptr, csr, embA,     outFirst, emb0, outSecond, N, 0);
  // layer 3: in=outFirst  -> embA     ; acc += e3
  lgc_propagate<<<PB, TB, 0, stream>>>(row_ptr, csr, outFirst, embA,     emb0, outSecond, N, 0);

  const long long total = (long long)N * LGC_D;
  lgc_finalize<<<(int)((total + TB - 1) / TB), TB, 0, stream>>>(emb0, outFirst, outSecond, total);
}